// TCN_82523501625770
// MI455X (gfx1250) — compile-verified
//
#include <hip/hip_runtime.h>

// TCN tail-only fused kernel for MI455X (gfx1250, wave32).
// Only the last output timestep is used by the reference, so only the last
// 5 input timesteps matter. Three small GEMMs, all on V_WMMA_F32_16X16X4_F32
// (exact fp32 matrix math -> matches the fp32 reference numerics).

typedef __attribute__((ext_vector_type(2))) float v2f;
typedef __attribute__((ext_vector_type(8))) float v8f;

#define B_    16
#define S_    16384
#define CIN   64
#define H_    128
#define OUT_  64
#define KW    3
#define TAIL  (S_ - 5)   // first input timestep that matters: 16379

__global__ __launch_bounds__(768)
void tcn_tail_wmma(const float* __restrict__ x,
                   const float* __restrict__ w1, const float* __restrict__ b1,
                   const float* __restrict__ w2, const float* __restrict__ b2,
                   const float* __restrict__ fw, const float* __restrict__ fb,
                   float* __restrict__ out)
{
    __shared__ float h1[48 * H_];   // rows m = b*3 + j  (j = tail timestep 0..2)
    __shared__ float h2[16 * H_];   // rows = batch

    const int tid  = threadIdx.x;
    const int wave = tid >> 5;          // 0..23
    const int lm   = tid & 15;          // lane mod 16
    const int half = (tid >> 4) & 1;    // lane group (K/M split per ISA layout)

    // ---------------- Step 1: h1 = relu(Xwin(48x192) * W1^T(192x128) + b1) ----
    {
        const int mtile = wave >> 3;            // 0..2
        const int ntile = wave & 7;             // 0..7
        const int m = mtile * 16 + lm;          // A-frag row: (b,j) pair
        const int b = m / 3;
        const int j = m - 3 * b;
        const int n = ntile * 16 + lm;          // B-frag col: hidden unit
        v8f acc = {0.f, 0.f, 0.f, 0.f, 0.f, 0.f, 0.f, 0.f};
        for (int t = 0; t < 48; ++t) {          // K = 192 = 48 * 4
            const int k0 = t * 4;
            v2f av, bv;
#pragma unroll
            for (int v = 0; v < 2; ++v) {
                const int i  = k0 + 2 * half + v;   // reduction idx = c*3 + k
                const int c  = i / 3;
                const int kk = i - 3 * c;
                av[v] = x[(size_t)(b * S_ + TAIL + j + kk) * CIN + c];
                bv[v] = w1[n * (CIN * KW) + i];
            }
            acc = __builtin_amdgcn_wmma_f32_16x16x4_f32(
                      false, av, false, bv, (short)0, acc, false, false);
        }
#pragma unroll
        for (int v = 0; v < 8; ++v) {
            const int row = mtile * 16 + v + 8 * half;
            const int h   = ntile * 16 + lm;
            h1[row * H_ + h] = fmaxf(acc[v] + b1[h], 0.f);
        }
    }
    __syncthreads();

    // ---------------- Step 2: h2 = relu(H1win(16x384) * W2^T(384x128) + b2) ---
    if (wave < 8) {
        const int n = wave * 16 + lm;           // hidden unit
        v8f acc = {0.f, 0.f, 0.f, 0.f, 0.f, 0.f, 0.f, 0.f};
        for (int t = 0; t < 96; ++t) {          // K = 384 = 96 * 4
            const int k0 = t * 4;
            v2f av, bv;
#pragma unroll
            for (int v = 0; v < 2; ++v) {
                const int i  = k0 + 2 * half + v;   // = c*3 + k
                const int c  = i / 3;
                const int kk = i - 3 * c;
                av[v] = h1[(lm * 3 + kk) * H_ + c]; // A row = batch = lm
                bv[v] = w2[n * (H_ * KW) + i];
            }
            acc = __builtin_amdgcn_wmma_f32_16x16x4_f32(
                      false, av, false, bv, (short)0, acc, false, false);
        }
#pragma unroll
        for (int v = 0; v < 8; ++v) {
            const int bi = v + 8 * half;        // batch
            const int h  = wave * 16 + lm;
            h2[bi * H_ + h] = fmaxf(acc[v] + b2[h], 0.f);
        }
    }
    __syncthreads();

    // ---------------- Step 3: out = H2(16x128) * FC^T(128x64) + fb ------------
    if (wave < 4) {
        const int n = wave * 16 + lm;           // output unit
        v8f acc = {0.f, 0.f, 0.f, 0.f, 0.f, 0.f, 0.f, 0.f};
        for (int t = 0; t < 32; ++t) {          // K = 128 = 32 * 4
            const int k0 = t * 4;
            v2f av, bv;
#pragma unroll
            for (int v = 0; v < 2; ++v) {
                const int i = k0 + 2 * half + v;    // hidden idx
                av[v] = h2[lm * H_ + i];            // A row = batch = lm
                bv[v] = fw[n * H_ + i];
            }
            acc = __builtin_amdgcn_wmma_f32_16x16x4_f32(
                      false, av, false, bv, (short)0, acc, false, false);
        }
#pragma unroll
        for (int v = 0; v < 8; ++v) {
            const int bi = v + 8 * half;        // batch
            out[bi * OUT_ + n] = acc[v] + fb[n];
        }
    }
}

extern "C" void kernel_launch(void* const* d_in, const int* in_sizes, int n_in,
                              void* d_out, int out_size, void* d_ws, size_t ws_size,
                              hipStream_t stream)
{
    const float* x  = (const float*)d_in[0];
    const float* w1 = (const float*)d_in[1];
    const float* b1 = (const float*)d_in[2];
    const float* w2 = (const float*)d_in[3];
    const float* b2 = (const float*)d_in[4];
    const float* fw = (const float*)d_in[5];
    const float* fb = (const float*)d_in[6];
    float* out = (float*)d_out;

    tcn_tail_wmma<<<1, 768, 0, stream>>>(x, w1, b1, w2, b2, fw, fb, out);
}